// GCN_88648124990615
// MI455X (gfx1250) — compile-verified
//
#include <hip/hip_runtime.h>
#include <hip/hip_bf16.h>

#define NGRAPHS 4096
#define HDIM 128

typedef __attribute__((ext_vector_type(2))) float v2f;
typedef __attribute__((ext_vector_type(8))) float v8f;

// ---------------- degree / norm ----------------
__global__ void k_deg_init(float* deg, int n) {
  int i = blockIdx.x * blockDim.x + threadIdx.x;
  if (i < n) deg[i] = 1.0f;  // self-loop contributes 1 to every node's in-degree
}
__global__ void k_deg_edges(const int* __restrict__ dst, float* deg, int ne) {
  int e = blockIdx.x * blockDim.x + threadIdx.x;
  if (e < ne) unsafeAtomicAdd(&deg[dst[e]], 1.0f);
}
__global__ void k_deg_rsqrt(float* deg, int n) {
  int i = blockIdx.x * blockDim.x + threadIdx.x;
  if (i < n) deg[i] = rsqrtf(deg[i]);  // deg >= 1 always (self loops)
}

// ---------------- WMMA GEMM: out[N,128] = H[N,K] @ W[K,128] ----------------
// GATHER=true: H is the embedding table, row index comes from x[] (fused gather).
// One wave computes a 16-row x 128-col tile via V_WMMA_F32_16X16X4_F32.
template <int K, bool GATHER>
__global__ void k_gemm(const float* __restrict__ H, const int* __restrict__ x,
                       const float* __restrict__ W, float* __restrict__ out, int n) {
  __shared__ float Wlds[K * HDIM];
  const int tid = threadIdx.x;
  for (int i = tid; i < K * HDIM; i += blockDim.x) Wlds[i] = W[i];
  __syncthreads();

  const int lane = tid & 31;
  const int wave = tid >> 5;
  const int r0 = (blockIdx.x * (blockDim.x >> 5) + wave) * 16;
  if (r0 >= n) return;                 // wave-uniform: EXEC stays all-1s

  const int mrow = lane & 15;          // M (for A) / N (for B,C,D) within tile
  const int kh   = (lane >> 4) << 1;   // lane half selects K pair {0,1} vs {2,3}

  if (r0 + 16 <= n) {
    const float* arow;
    if (GATHER) arow = H + (long long)x[r0 + mrow] * K;
    else        arow = H + (long long)(r0 + mrow) * K;

    v8f acc[8] = {};
    for (int k = 0; k < K; k += 4) {
      v2f a;                            // A 16x4 f32: V0=K+kh, V1=K+kh+1
      a.x = arow[k + kh];
      a.y = arow[k + kh + 1];
#pragma unroll
      for (int t = 0; t < 8; ++t) {     // 8 column tiles of 16
        v2f b;                          // B 4x16 f32: V0=row k+kh, V1=row k+kh+1
        b.x = Wlds[(k + kh) * HDIM + t * 16 + mrow];
        b.y = Wlds[(k + kh + 1) * HDIM + t * 16 + mrow];
        acc[t] = __builtin_amdgcn_wmma_f32_16x16x4_f32(
            false, a, false, b, (short)0, acc[t], false, false);
      }
    }
    // D layout: VGPR r -> row r (lanes 0-15) / row r+8 (lanes 16-31), col = lane&15
    const int rbase = r0 + ((lane >> 4) << 3);
#pragma unroll
    for (int t = 0; t < 8; ++t)
#pragma unroll
      for (int r = 0; r < 8; ++r)
        out[(long long)(rbase + r) * HDIM + t * 16 + mrow] = acc[t][r];
  } else {
    // generic tail (unused for N=300000 which is a multiple of 16)
    for (int row = r0 + lane; row < n; row += 32) {
      const float* arow;
      if (GATHER) arow = H + (long long)x[row] * K;
      else        arow = H + (long long)row * K;
      for (int c = 0; c < HDIM; ++c) {
        float s = 0.f;
        for (int k = 0; k < K; ++k) s += arow[k] * Wlds[k * HDIM + c];
        out[(long long)row * HDIM + c] = s;
      }
    }
  }
}

// ---------------- self-loop + bias init: acc = XW*dinv^2 + b ----------------
__global__ void k_init_acc(const float* __restrict__ XW, const float* __restrict__ dinv,
                           const float* __restrict__ bias, float* __restrict__ acc, int n) {
  long long idx = (long long)blockIdx.x * blockDim.x + threadIdx.x;
  int i = (int)(idx >> 5);
  int c = ((int)idx & 31) * 4;
  if (i >= n) return;
  float w = dinv[i] * dinv[i];
  const float4 v = *(const float4*)(XW + (long long)i * HDIM + c);
  const float4 b = *(const float4*)(bias + c);
  float4 r;
  r.x = v.x * w + b.x; r.y = v.y * w + b.y;
  r.z = v.z * w + b.z; r.w = v.w * w + b.w;
  *(float4*)(acc + (long long)i * HDIM + c) = r;
}

// ---------------- edge scatter: acc[dst] += XW[src] * dinv[s]*dinv[d] ----------------
__global__ void k_edge_scatter(const int* __restrict__ src, const int* __restrict__ dst,
                               const float* __restrict__ dinv, const float* __restrict__ XW,
                               float* __restrict__ acc, int ne) {
  long long idx = (long long)blockIdx.x * blockDim.x + threadIdx.x;
  int e = (int)(idx >> 5);
  int c = ((int)idx & 31) * 4;
  if (e >= ne) return;
  int s = src[e], d = dst[e];
  float w = dinv[s] * dinv[d];
  const float4 v = *(const float4*)(XW + (long long)s * HDIM + c);
  float* p = acc + (long long)d * HDIM + c;
  unsafeAtomicAdd(p + 0, v.x * w);
  unsafeAtomicAdd(p + 1, v.y * w);
  unsafeAtomicAdd(p + 2, v.z * w);
  unsafeAtomicAdd(p + 3, v.w * w);
}

// ---------------- relu in place ----------------
__global__ void k_relu(float* a, long long n4) {
  long long i = (long long)blockIdx.x * blockDim.x + threadIdx.x;
  if (i >= n4) return;
  float4* p = (float4*)a + i;
  float4 v = *p;
  v.x = fmaxf(v.x, 0.f); v.y = fmaxf(v.y, 0.f);
  v.z = fmaxf(v.z, 0.f); v.w = fmaxf(v.w, 0.f);
  *p = v;
}

// ---------------- mean-pool scatter ----------------
__global__ void k_pool(const int* __restrict__ batch, const float* __restrict__ h,
                       float* __restrict__ pooled, float* __restrict__ counts, int n) {
  long long idx = (long long)blockIdx.x * blockDim.x + threadIdx.x;
  int i = (int)(idx >> 5);
  int c = ((int)idx & 31) * 4;
  if (i >= n) return;
  int g = batch[i];
  const float4 v = *(const float4*)(h + (long long)i * HDIM + c);
  float* p = pooled + (long long)g * HDIM + c;
  unsafeAtomicAdd(p + 0, v.x);
  unsafeAtomicAdd(p + 1, v.y);
  unsafeAtomicAdd(p + 2, v.z);
  unsafeAtomicAdd(p + 3, v.w);
  if (c == 0) unsafeAtomicAdd(&counts[g], 1.0f);
}

// ---------------- classifier: out = (pooled/count) @ Wl + bl ----------------
__global__ void k_final(const float* __restrict__ pooled, const float* __restrict__ counts,
                        const float* __restrict__ Wl, const float* __restrict__ bl,
                        float* __restrict__ out, int ngraphs) {
  int idx = blockIdx.x * blockDim.x + threadIdx.x;
  int g = idx / 10, j = idx % 10;
  if (g >= ngraphs) return;
  float inv = 1.0f / fmaxf(counts[g], 1.0f);
  float s = bl[j];
  const float* pr = pooled + (long long)g * HDIM;
  for (int c = 0; c < HDIM; ++c) s += pr[c] * inv * Wl[c * 10 + j];
  out[idx] = s;
}

extern "C" void kernel_launch(void* const* d_in, const int* in_sizes, int n_in,
                              void* d_out, int out_size, void* d_ws, size_t ws_size,
                              hipStream_t stream) {
  const int n  = in_sizes[0];
  const int ne = in_sizes[1] / 2;
  const int*   x     = (const int*)d_in[0];
  const int*   src   = (const int*)d_in[1];
  const int*   dst   = src + ne;
  const int*   batch = (const int*)d_in[2];
  const float* emb   = (const float*)d_in[4];
  const float* W1    = (const float*)d_in[5];
  const float* b1    = (const float*)d_in[6];
  const float* W2    = (const float*)d_in[7];
  const float* b2    = (const float*)d_in[8];
  const float* Wl    = (const float*)d_in[9];
  const float* bl    = (const float*)d_in[10];
  float* out = (float*)d_out;

  char* ws = (char*)d_ws;
  auto carve = [&](size_t bytes) {
    char* p = ws;
    ws += (bytes + 255) & ~(size_t)255;
    return p;
  };
  float* dinv   = (float*)carve((size_t)n * 4);
  float* XW     = (float*)carve((size_t)n * HDIM * 4);
  float* ACC    = (float*)carve((size_t)n * HDIM * 4);
  float* pooled = (float*)carve((size_t)NGRAPHS * HDIM * 4);
  float* counts = (float*)carve((size_t)NGRAPHS * 4);

  const int T = 256;
  const long long nc = (long long)n * 32;   // (n*128)/4 float4 lanes
  const long long ec = (long long)ne * 32;  // 32 lanes per edge
  const int waves   = (n + 15) / 16;
  const int gblocks = (waves + 3) / 4;      // 128-thread blocks = 4 waves

  // degrees -> dinv (in place)
  k_deg_init<<<(n + T - 1) / T, T, 0, stream>>>(dinv, n);
  k_deg_edges<<<(ne + T - 1) / T, T, 0, stream>>>(dst, dinv, ne);
  k_deg_rsqrt<<<(n + T - 1) / T, T, 0, stream>>>(dinv, n);

  // layer 1: fused embedding-gather GEMM (K=64) + conv + relu
  k_gemm<64, true><<<gblocks, 128, 0, stream>>>(emb, x, W1, XW, n);
  k_init_acc<<<(unsigned)((nc + T - 1) / T), T, 0, stream>>>(XW, dinv, b1, ACC, n);
  k_edge_scatter<<<(unsigned)((ec + T - 1) / T), T, 0, stream>>>(src, dst, dinv, XW, ACC, ne);
  k_relu<<<(unsigned)((nc + T - 1) / T), T, 0, stream>>>(ACC, nc);

  // layer 2: GEMM (K=128) + conv + relu (ACC reused after GEMM consumes it)
  k_gemm<128, false><<<gblocks, 128, 0, stream>>>(ACC, nullptr, W2, XW, n);
  k_init_acc<<<(unsigned)((nc + T - 1) / T), T, 0, stream>>>(XW, dinv, b2, ACC, n);
  k_edge_scatter<<<(unsigned)((ec + T - 1) / T), T, 0, stream>>>(src, dst, dinv, XW, ACC, ne);
  k_relu<<<(unsigned)((nc + T - 1) / T), T, 0, stream>>>(ACC, nc);

  // mean pool + classifier
  hipMemsetAsync(pooled, 0, (size_t)NGRAPHS * HDIM * 4, stream);
  hipMemsetAsync(counts, 0, (size_t)NGRAPHS * 4, stream);
  k_pool<<<(unsigned)((nc + T - 1) / T), T, 0, stream>>>(batch, ACC, pooled, counts, n);
  k_final<<<(NGRAPHS * 10 + T - 1) / T, T, 0, stream>>>(pooled, counts, Wl, bl, out, NGRAPHS);
}